// GeneralContactLoss_29042568855593
// MI455X (gfx1250) — compile-verified
//
#include <hip/hip_runtime.h>
#include <math.h>

typedef __attribute__((ext_vector_type(2))) float v2f;
typedef __attribute__((ext_vector_type(8))) float v8f;

#define BT 32      // B*T frames
#define NV 10475   // high-res verts
#define NW 32      // mouth verts
#define VL 500     // low-res verts
#define FL 996     // low-res faces
#define VP 512     // padded low-res count (32 tiles of 16)

// ---- ws layout (bytes) ----
// XY       : float [2][BT][VP][4]   @ 0        (524288 B)  (x,y,z,|v|^2); s=0 -> v1, s=1 -> v2
// interior : int   [2][BT][VP]      @ 524288   (131072 B)
// crit     : int   [2][BT]          @ 655360   (256 B)
// fsum     : float [2][BT]          @ 655616   (256 B)

__global__ void prep_kernel(const float* __restrict__ v1, const float* __restrict__ v2,
                            const int* __restrict__ vert_ids, const int* __restrict__ lowres_vid,
                            float* __restrict__ XY, int* __restrict__ crit, float* __restrict__ fsum)
{
    const int f = blockIdx.x;
    const int tid = threadIdx.x;
    __shared__ float mouth[2][3];
    if (tid < 2) {
        const float* src = tid ? v2 : v1;
        const size_t base = (size_t)f * NV * 3;
        float ax = 0.f, ay = 0.f, az = 0.f;
        for (int w = 0; w < NW; ++w) {
            const float* p = src + base + (size_t)vert_ids[w] * 3;
            ax += p[0]; ay += p[1]; az += p[2];
        }
        mouth[tid][0] = ax * (1.0f / NW);
        mouth[tid][1] = ay * (1.0f / NW);
        mouth[tid][2] = az * (1.0f / NW);
    }
    if (tid == 2) { // zero accumulators (ws is poisoned; must re-init every call)
        crit[f] = 0; crit[BT + f] = 0;
        fsum[f] = 0.f; fsum[BT + f] = 0.f;
    }
    __syncthreads();
    for (int i = tid; i < VP; i += blockDim.x) {
        for (int s = 0; s < 2; ++s) {
            float x, y, z;
            if (i < VL) {
                int vid = lowres_vid[i];
                if (vid == NV) { x = mouth[s][0]; y = mouth[s][1]; z = mouth[s][2]; }
                else {
                    const float* src = s ? v2 : v1;
                    const float* p = src + (size_t)f * NV * 3 + (size_t)vid * 3;
                    x = p[0]; y = p[1]; z = p[2];
                }
            } else { x = 1e18f; y = 1e18f; z = 1e18f; } // padded: huge distance, f32-safe
            float n = x * x + y * y + z * z;
            float* o = XY + (((size_t)s * BT + f) * VP + i) * 4;
            o[0] = x; o[1] = y; o[2] = z; o[3] = n;
        }
    }
}

__global__ void winding_kernel(const float* __restrict__ XY, const int* __restrict__ faces,
                               int* __restrict__ interior, int* __restrict__ crit)
{
    const int f = blockIdx.x;
    const int dir = blockIdx.y;   // 0: points=v1, tris=v2 ; 1: points=v2, tris=v1
    const int chunk = blockIdx.z;
    const int tid = threadIdx.x;
    __shared__ float4 tv[VP];
    __shared__ int fidx[FL * 3];
    const float* src = XY + (((size_t)(1 - dir) * BT + f) * VP) * 4;
    for (int i = tid; i < VP; i += blockDim.x)
        tv[i] = make_float4(src[i * 4 + 0], src[i * 4 + 1], src[i * 4 + 2], 0.f);
    for (int i = tid; i < FL * 3; i += blockDim.x) fidx[i] = faces[i];
    __syncthreads();
    const int p = chunk * 256 + tid;
    if (p >= VP) return;
    int flag = 0;
    if (p < VL) {
        const float* pp = XY + (((size_t)dir * BT + f) * VP + p) * 4;
        const float px = pp[0], py = pp[1], pz = pp[2];
        float acc = 0.f;
        for (int t = 0; t < FL; ++t) {
            float4 A4 = tv[fidx[3 * t + 0]];
            float4 B4 = tv[fidx[3 * t + 1]];
            float4 C4 = tv[fidx[3 * t + 2]];
            float ax = A4.x - px, ay = A4.y - py, az = A4.z - pz;
            float bx = B4.x - px, by = B4.y - py, bz = B4.z - pz;
            float cx = C4.x - px, cy = C4.y - py, cz = C4.z - pz;
            float la = sqrtf(ax * ax + ay * ay + az * az);
            float lb = sqrtf(bx * bx + by * by + bz * bz);
            float lc = sqrtf(cx * cx + cy * cy + cz * cz);
            float kx = by * cz - bz * cy;
            float ky = bz * cx - bx * cz;
            float kz = bx * cy - by * cx;
            float det = ax * kx + ay * ky + az * kz;
            float ab = ax * bx + ay * by + az * bz;
            float bc = bx * cx + by * cy + bz * cz;
            float ca = cx * ax + cy * ay + cz * az;
            float den = la * lb * lc + ab * lc + bc * la + ca * lb;
            acc += atan2f(det, den);
        }
        // solid = 2*atan2 ; wn = sum(solid)/(4*pi) = acc/(2*pi)
        float wn = acc * 0.15915494309189535f;
        flag = (wn >= 0.99f) ? 1 : 0;
    }
    interior[((size_t)dir * BT + f) * VP + p] = flag;
    if (flag) atomicOr(&crit[dir * BT + f], 1);
}

__global__ void __launch_bounds__(1024)
mindist_kernel(const float* __restrict__ XY, const int* __restrict__ interior,
               float* __restrict__ fsum)
{
    const int f = blockIdx.x;
    const int dir = blockIdx.y;
    const int tid = threadIdx.x;
    __shared__ float4 Xs[VP];
    __shared__ float4 Ys[VP];
    const float4* Xg = (const float4*)(XY + (((size_t)dir * BT + f) * VP) * 4);
    const float4* Yg = (const float4*)(XY + (((size_t)(1 - dir) * BT + f) * VP) * 4);
    if (tid < VP) Xs[tid] = Xg[tid];
    else          Ys[tid - VP] = Yg[tid - VP];
    __syncthreads();

    const int wave = tid >> 5;
    const int lane = tid & 31;
    const int hi   = lane >> 4;   // lane half selects K pair {0,1} vs {2,3}
    const int l16  = lane & 15;
    const int m0   = wave * 16;   // row tile for this wave

    // A tile (16x4): reg0 = K0 (lanes 0-15) / K2 (lanes 16-31); reg1 = K1 / K3(=0)
    float4 xa = Xs[m0 + l16];
    v2f a;
    a.x = hi ? xa.z : xa.x;
    a.y = hi ? 0.f : xa.y;

    float nx[8];
#pragma unroll
    for (int r = 0; r < 8; ++r) nx[r] = Xs[m0 + r + (hi ? 8 : 0)].w; // |x_m|^2 per D-reg

    float mn[8];
#pragma unroll
    for (int r = 0; r < 8; ++r) mn[r] = 3.0e38f;

    for (int j0 = 0; j0 < VP; j0 += 16) {
        float4 yb = Ys[j0 + l16];
        v2f b;                         // B tile (4x16), K rows striped like A
        b.x = hi ? yb.z : yb.x;
        b.y = hi ? 0.f : yb.y;
        float nyl = yb.w;              // |y_n|^2, n = j0 + (lane&15)
        v8f c = {};
        c = __builtin_amdgcn_wmma_f32_16x16x4_f32(false, a, false, b,
                                                  (short)0, c, false, false);
#pragma unroll
        for (int r = 0; r < 8; ++r) {
            float d2 = nx[r] + nyl - 2.0f * c[r];
            mn[r] = fminf(mn[r], d2);
        }
    }

    // min over N: xor-shuffles with offsets <16 stay within each 16-lane half
#pragma unroll
    for (int r = 0; r < 8; ++r) {
        float m = mn[r];
        m = fminf(m, __shfl_xor(m, 8, 32));
        m = fminf(m, __shfl_xor(m, 4, 32));
        m = fminf(m, __shfl_xor(m, 2, 32));
        m = fminf(m, __shfl_xor(m, 1, 32));
        mn[r] = m;
    }

    if (l16 == 0) { // lane 0: rows m0..m0+7 ; lane 16: rows m0+8..m0+15
        const int* intr = interior + ((size_t)dir * BT + f) * VP;
        float lsum = 0.f;
#pragma unroll
        for (int r = 0; r < 8; ++r) {
            int row = m0 + r + (hi ? 8 : 0);
            if (row < VL && intr[row]) {
                float d2 = fmaxf(mn[r], 1e-12f);
                float d = sqrtf(d2) * 1000.0f;
                lsum += d * d;
            }
        }
        atomicAdd(&fsum[dir * BT + f], lsum);
    }
}

__global__ void finalize_kernel(const float* __restrict__ mask, const float* __restrict__ tmask,
                                const int* __restrict__ crit, const float* __restrict__ fsum,
                                float* __restrict__ out)
{
    const int f = threadIdx.x; // 0..31, one wave
    float m  = mask[f];
    float tm = tmask[f >> 3];  // T = 8
    float cf = (crit[f] != 0 && crit[BT + f] != 0) ? 1.f : 0.f;
    float factor = cf * m * tm;
    float num = factor * (fsum[f] + fsum[BT + f]);
    float den = factor;
    for (int off = 16; off; off >>= 1) {
        num += __shfl_xor(num, off, 32);
        den += __shfl_xor(den, off, 32);
    }
    if (f == 0) out[0] = num / (den + 1e-7f);
}

extern "C" void kernel_launch(void* const* d_in, const int* in_sizes, int n_in,
                              void* d_out, int out_size, void* d_ws, size_t ws_size,
                              hipStream_t stream)
{
    const float* v1         = (const float*)d_in[0];
    const float* v2         = (const float*)d_in[1];
    const float* mask       = (const float*)d_in[2];
    const float* tmask      = (const float*)d_in[3];
    const int*   vert_ids   = (const int*)d_in[4];
    const int*   lowres_vid = (const int*)d_in[5];
    const int*   faces      = (const int*)d_in[6];

    char* ws = (char*)d_ws;
    float* XY      = (float*)(ws);
    int*   interior= (int*)  (ws + 524288);
    int*   crit    = (int*)  (ws + 655360);
    float* fsum    = (float*)(ws + 655616);

    prep_kernel   <<<dim3(BT),        dim3(256),  0, stream>>>(v1, v2, vert_ids, lowres_vid, XY, crit, fsum);
    winding_kernel<<<dim3(BT, 2, 2),  dim3(256),  0, stream>>>(XY, faces, interior, crit);
    mindist_kernel<<<dim3(BT, 2),     dim3(1024), 0, stream>>>(XY, interior, fsum);
    finalize_kernel<<<dim3(1),        dim3(32),   0, stream>>>(mask, tmask, crit, fsum, (float*)d_out);
}